// SSD_64398739636704
// MI455X (gfx1250) — compile-verified
//
#include <hip/hip_runtime.h>
#include <math.h>

// ---------------- problem constants (SSD300 postprocess) ----------------
#define NIMG   32
#define ANUM   8732
#define NCLS   91
#define NFG    90
#define KPRE   1000
#define KPOST  200
#define SCORE_THR 0.01f
#define IOU_THR   0.45f
#define IMGSZ     300.0f
#define XCLIP     4.1351665567423560f   // log(1000/16)
#define CAND_CAP  65536
#define NBINS     4096
#define NSORT     4096
#define BIN_BASE  0x3C000000u           // float bits of 0.0078125 (< 0.01)
#define BIN_SHIFT 14

typedef float v2f __attribute__((ext_vector_type(2)));
typedef float v8f __attribute__((ext_vector_type(8)));
typedef unsigned long long u64;
typedef unsigned int u32;

// ---------------- kernel 0: zero histograms + counters ----------------
__global__ void k_init(u32* __restrict__ hist, int* __restrict__ cnt) {
    int i = blockIdx.x * blockDim.x + threadIdx.x;
    if (i < NIMG * NBINS) hist[i] = 0u;
    if (i < NIMG) cnt[i] = 0;
}

// ---------------- kernel 1: softmax + candidate scan ----------------
// One wave owns 16 anchors. Row-sums of exp(x - rowmax) computed with
// V_WMMA_F32_16X16X4_F32 (A = 16x4 exp tile, B = all-ones, C accumulates
// over 23 chunks of 4 classes). D layout: lane0 comp v = rowsum(v),
// lane16 comp v = rowsum(v+8)  (ISA 7.12.2, 32-bit C/D 16x16).
__global__ __launch_bounds__(256)
void k_softmax_scan(const float* __restrict__ cls,
                    u32* __restrict__ hist,            // [NIMG][NBINS]
                    int* __restrict__ cnt,             // [NIMG]
                    u64* __restrict__ cand)            // [NIMG][CAND_CAP]
{
    __shared__ float s_sum[8][16];
    const int blocksPerImg = (ANUM + 127) / 128;       // 69
    int img  = blockIdx.x / blocksPerImg;
    int blk  = blockIdx.x % blocksPerImg;
    int w    = threadIdx.x >> 5;
    int lane = threadIdx.x & 31;
    int row  = lane & 15;
    int hi   = lane >> 4;                              // 0: K=0,1  1: K=2,3
    int base = blk * 128 + w * 16;
    int anchor = base + row;
    int aclamp = (anchor < ANUM) ? anchor : (ANUM - 1);
    const float* prow = cls + ((size_t)img * ANUM + aclamp) * NCLS;

    // phase 1: per-row max (lane pair L / L^16 covers all 91 classes)
    float m = -3.0e38f;
    #pragma unroll
    for (int t = 0; t < 23; ++t) {
        int c0 = 4 * t + 2 * hi;
        if (c0 < NCLS)     m = fmaxf(m, prow[c0]);
        if (c0 + 1 < NCLS) m = fmaxf(m, prow[c0 + 1]);
    }
    m = fmaxf(m, __shfl_xor(m, 16, 32));

    // phase 2: row sums of exp via WMMA (B = ones -> every D column = rowsum)
    v8f acc = {0.f, 0.f, 0.f, 0.f, 0.f, 0.f, 0.f, 0.f};
    v2f onesB; onesB.x = 1.0f; onesB.y = 1.0f;
    #pragma unroll
    for (int t = 0; t < 23; ++t) {
        int c0 = 4 * t + 2 * hi;
        v2f a;
        a.x = (c0     < NCLS) ? __expf(prow[c0]     - m) : 0.0f;
        a.y = (c0 + 1 < NCLS) ? __expf(prow[c0 + 1] - m) : 0.0f;
        acc = __builtin_amdgcn_wmma_f32_16x16x4_f32(
                  false, a, false, onesB, (short)0, acc, false, false);
    }
    if (lane == 0) {
        #pragma unroll
        for (int v = 0; v < 8; ++v) s_sum[w][v] = acc[v];
    }
    if (lane == 16) {
        #pragma unroll
        for (int v = 0; v < 8; ++v) s_sum[w][8 + v] = acc[v];
    }
    __syncthreads();

    // phase 3: emit candidates (classes 1..90; background dropped)
    if (anchor < ANUM) {
        float invs = 1.0f / s_sum[w][row];
        for (int c = 1 + hi; c < NCLS; c += 2) {
            float s = __expf(prow[c] - m) * invs;
            if (s > SCORE_THR) {
                u32 bits = __float_as_uint(s);
                int bin = (int)((bits - BIN_BASE) >> BIN_SHIFT);
                bin = bin < 0 ? 0 : (bin > NBINS - 1 ? NBINS - 1 : bin);
                atomicAdd(&hist[img * NBINS + bin], 1u);
                int slot = atomicAdd(&cnt[img], 1);
                if (slot < CAND_CAP) {
                    u32 idx = (u32)(anchor * NFG + (c - 1));
                    cand[(size_t)img * CAND_CAP + slot] =
                        ((u64)bits << 32) | (u64)(0xFFFFFFFFu - idx);
                }
            }
        }
    }
}

// ---------------- kernel 2: histogram suffix scan -> score cutoff ----------------
__global__ void k_cutoff(const u32* __restrict__ hist, const int* __restrict__ cnt,
                         u32* __restrict__ cutoff)
{
    int img = threadIdx.x;
    if (img >= NIMG) return;
    int total = cnt[img]; if (total > CAND_CAP) total = CAND_CAP;
    int target = total < KPRE ? total : KPRE;
    u32 cb = NBINS;                     // nothing selected -> cutoff above all
    if (target > 0) {
        int cum = 0;
        for (int b = NBINS - 1; b >= 0; --b) {
            cum += (int)hist[img * NBINS + b];
            if (cum >= target) { cb = (u32)b; break; }
        }
        if (cb == NBINS) cb = 0;        // defensive (cum must reach target)
    }
    cutoff[img] = BIN_BASE + (cb << BIN_SHIFT);
}

// ---------------- kernel 3: gather + bitonic top-1000 + box decode ----------------
__global__ __launch_bounds__(256)
void k_select_sort_decode(const u64* __restrict__ cand, const int* __restrict__ cnt,
                          const u32* __restrict__ cutoff,
                          const float* __restrict__ reg, const float* __restrict__ priors,
                          float* __restrict__ selScore, int* __restrict__ selCls,
                          float* __restrict__ boxes, float* __restrict__ boxesOff)
{
    __shared__ u64 key[NSORT];
    __shared__ int sn;
    int img = blockIdx.x;
    int tid = threadIdx.x;
    if (tid == 0) sn = 0;
    for (int i = tid; i < NSORT; i += 256) key[i] = 0ull;
    __syncthreads();

    int total = cnt[img]; if (total > CAND_CAP) total = CAND_CAP;
    u64 cutk = ((u64)cutoff[img]) << 32;
    for (int i = tid; i < total; i += 256) {
        u64 k = cand[(size_t)img * CAND_CAP + i];
        if (k >= cutk) {
            int slot = atomicAdd(&sn, 1);
            if (slot < NSORT) key[slot] = k;
        }
    }
    __syncthreads();

    // bitonic sort, descending; keys unique -> fully deterministic order
    for (int kk = 2; kk <= NSORT; kk <<= 1) {
        for (int j = kk >> 1; j > 0; j >>= 1) {
            for (int i = tid; i < NSORT; i += 256) {
                int ixj = i ^ j;
                if (ixj > i) {
                    bool desc = ((i & kk) == 0);
                    u64 a = key[i], b = key[ixj];
                    if ((a < b) == desc) { key[i] = b; key[ixj] = a; }
                }
            }
            __syncthreads();
        }
    }

    // decode top KPRE
    for (int r = tid; r < KPRE; r += 256) {
        u64 k = key[r];
        u32 bits = (u32)(k >> 32);
        size_t o1 = (size_t)img * KPRE + r;
        if (bits == 0u) {
            selScore[o1] = 0.0f; selCls[o1] = 0;
            #pragma unroll
            for (int q = 0; q < 4; ++q) { boxes[o1 * 4 + q] = 0.0f; boxesOff[o1 * 4 + q] = 0.0f; }
        } else {
            float score = __uint_as_float(bits);
            u32 idx = 0xFFFFFFFFu - (u32)(k & 0xFFFFFFFFull);
            int anchor = (int)(idx / NFG);
            int clsI = (int)(idx % NFG) + 1;
            const float* rg = reg + ((size_t)img * ANUM + anchor) * 4;
            const float* pr = priors + (size_t)anchor * 4;
            float cx = pr[0] + rg[0] * 0.1f * pr[2];
            float cy = pr[1] + rg[1] * 0.1f * pr[3];
            float bw = pr[2] * __expf(fminf(rg[2] * 0.2f, XCLIP));
            float bh = pr[3] * __expf(fminf(rg[3] * 0.2f, XCLIP));
            float x1 = fminf(fmaxf(cx - 0.5f * bw, 0.0f), IMGSZ);
            float y1 = fminf(fmaxf(cy - 0.5f * bh, 0.0f), IMGSZ);
            float x2 = fminf(fmaxf(cx + 0.5f * bw, 0.0f), IMGSZ);
            float y2 = fminf(fmaxf(cy + 0.5f * bh, 0.0f), IMGSZ);
            float off = (float)clsI * (IMGSZ + 1.0f);
            boxes[o1 * 4 + 0] = x1; boxes[o1 * 4 + 1] = y1;
            boxes[o1 * 4 + 2] = x2; boxes[o1 * 4 + 3] = y2;
            boxesOff[o1 * 4 + 0] = x1 + off; boxesOff[o1 * 4 + 1] = y1 + off;
            boxesOff[o1 * 4 + 2] = x2 + off; boxesOff[o1 * 4 + 3] = y2 + off;
            selScore[o1] = score; selCls[o1] = clsI;
        }
    }
}

// ---------------- kernel 4: greedy NMS, one workgroup per image ----------------
__global__ __launch_bounds__(256)
void k_nms(const float* __restrict__ boxesOff, const float* __restrict__ selScore,
           int* __restrict__ keepOut)
{
    __shared__ float4 bo[KPRE];
    __shared__ unsigned char kp[KPRE];
    int img = blockIdx.x, tid = threadIdx.x;
    for (int j = tid; j < KPRE; j += 256) {
        const float* p = boxesOff + ((size_t)img * KPRE + j) * 4;
        bo[j] = make_float4(p[0], p[1], p[2], p[3]);
        kp[j] = (selScore[(size_t)img * KPRE + j] > SCORE_THR) ? 1 : 0;
    }
    __syncthreads();
    for (int i = 0; i < KPRE; ++i) {
        if (kp[i]) {                               // uniform read
            float4 bi = bo[i];
            float ai = (bi.z - bi.x) * (bi.w - bi.y);
            for (int j = i + 1 + tid; j < KPRE; j += 256) {
                if (kp[j]) {
                    float4 bj = bo[j];
                    float lx = fmaxf(bi.x, bj.x), ly = fmaxf(bi.y, bj.y);
                    float rx = fminf(bi.z, bj.z), ry = fminf(bi.w, bj.w);
                    float iw = fmaxf(rx - lx, 0.0f), ih = fmaxf(ry - ly, 0.0f);
                    float inter = iw * ih;
                    float aj = (bj.z - bj.x) * (bj.w - bj.y);
                    float iou = inter / (ai + aj - inter + 1e-9f);
                    if (iou > IOU_THR) kp[j] = 0;
                }
            }
        }
        __syncthreads();
    }
    for (int j = tid; j < KPRE; j += 256) keepOut[(size_t)img * KPRE + j] = kp[j];
}

// ---------------- kernel 5: wave32 ballot compaction into keep[:200] ----------------
__global__ __launch_bounds__(32)
void k_finalize(const int* __restrict__ keep, const float* __restrict__ selScore,
                const int* __restrict__ selCls, const float* __restrict__ boxes,
                float* __restrict__ out)
{
    int img = blockIdx.x, lane = threadIdx.x;
    float* b2 = out + (size_t)img * KPOST * 4;
    float* s2 = out + (size_t)NIMG * KPOST * 4 + (size_t)img * KPOST;
    float* c2 = out + (size_t)NIMG * KPOST * 5 + (size_t)img * KPOST;
    int outn = 0;
    for (int basei = 0; basei < KPRE; basei += 32) {
        int r = basei + lane;
        bool k = (r < KPRE) && (keep[(size_t)img * KPRE + r] != 0);
        unsigned mask = (unsigned)__ballot(k ? 1 : 0);      // wave32: low 32 bits
        int pre = __popc(mask & ((1u << lane) - 1u));
        int pos = outn + pre;
        if (k && pos < KPOST) {
            const float* bp = boxes + ((size_t)img * KPRE + r) * 4;
            b2[pos * 4 + 0] = bp[0]; b2[pos * 4 + 1] = bp[1];
            b2[pos * 4 + 2] = bp[2]; b2[pos * 4 + 3] = bp[3];
            s2[pos] = selScore[(size_t)img * KPRE + r];
            c2[pos] = (float)selCls[(size_t)img * KPRE + r];
        }
        outn += __popc(mask);
        if (outn >= KPOST) break;                           // uniform
    }
    int start = outn < KPOST ? outn : KPOST;
    for (int p = start + lane; p < KPOST; p += 32) {
        b2[p * 4 + 0] = 0.0f; b2[p * 4 + 1] = 0.0f;
        b2[p * 4 + 2] = 0.0f; b2[p * 4 + 3] = 0.0f;
        s2[p] = 0.0f; c2[p] = 0.0f;
    }
}

// ---------------- host launch ----------------
extern "C" void kernel_launch(void* const* d_in, const int* in_sizes, int n_in,
                              void* d_out, int out_size, void* d_ws, size_t ws_size,
                              hipStream_t stream) {
    (void)in_sizes; (void)n_in; (void)out_size; (void)ws_size;
    const float* cls    = (const float*)d_in[0];   // [32,8732,91]
    const float* reg    = (const float*)d_in[1];   // [32,8732,4]
    const float* priors = (const float*)d_in[2];   // [8732,4]
    float* out = (float*)d_out;                    // b2(25600) | s2(6400) | c2(6400)

    char* ws = (char*)d_ws;
    size_t off = 0;
    auto take = [&](size_t bytes) { char* p = ws + off; off += (bytes + 255) & ~(size_t)255; return p; };
    u64* cand      = (u64*)take((size_t)NIMG * CAND_CAP * sizeof(u64));
    int* cnt       = (int*)take((size_t)NIMG * sizeof(int));
    u32* hist      = (u32*)take((size_t)NIMG * NBINS * sizeof(u32));
    u32* cutoff    = (u32*)take((size_t)NIMG * sizeof(u32));
    float* selScore= (float*)take((size_t)NIMG * KPRE * sizeof(float));
    int*   selCls  = (int*)take((size_t)NIMG * KPRE * sizeof(int));
    float* boxes   = (float*)take((size_t)NIMG * KPRE * 4 * sizeof(float));
    float* boxesOff= (float*)take((size_t)NIMG * KPRE * 4 * sizeof(float));
    int*   keep    = (int*)take((size_t)NIMG * KPRE * sizeof(int));

    int initN = NIMG * NBINS;
    k_init<<<(initN + 255) / 256, 256, 0, stream>>>(hist, cnt);

    const int blocksPerImg = (ANUM + 127) / 128;   // 69
    k_softmax_scan<<<NIMG * blocksPerImg, 256, 0, stream>>>(cls, hist, cnt, cand);

    k_cutoff<<<1, 32, 0, stream>>>(hist, cnt, cutoff);

    k_select_sort_decode<<<NIMG, 256, 0, stream>>>(cand, cnt, cutoff, reg, priors,
                                                   selScore, selCls, boxes, boxesOff);

    k_nms<<<NIMG, 256, 0, stream>>>(boxesOff, selScore, keep);

    k_finalize<<<NIMG, 32, 0, stream>>>(keep, selScore, selCls, boxes, out);
}